// FusedTensorProductOp3_55808805044384
// MI455X (gfx1250) — compile-verified
//
#include <hip/hip_runtime.h>

// ---------------------------------------------------------------------------
// Fused segmented tensor product (UUVV), MI455X / gfx1250, wave32 + WMMA.
//
//   out[b, j*128+v] = sum_i sum_u COEFS[i][j] * in0[b, i*128+u]
//                                             * in1[b, ((i*4+j)*128+u)*128+v]
//
// Memory-bound (0.5 FLOP/byte): in1 (512 MB) streamed once, non-temporal.
// WMMA F32_16x16x4 used as a replicated-row GEMV engine:
//   A[M][k] = c * x0[u+k]  (same for all 16 rows M)
//   B[k][n] = x1[u+k][v0+n]
//   D accumulates over i and u; row 0 of D (acc[0], lanes 0..15) is the result.
//
// A-operand layout (32-bit A, 16x4): lanes 0-15 hold K=0,1 in VGPR0,1;
// lanes 16-31 hold K=2,3.  B-operand (4x16): VGPR0 = row K=2h, VGPR1 = row
// K=2h+1 for lane-half h, column N = lane&15.
// ---------------------------------------------------------------------------

typedef __attribute__((ext_vector_type(2))) float v2f;
typedef __attribute__((ext_vector_type(8))) float v8f;

#define NU 4
#define NV 4
#define UDIM 128
#define VDIM 128

__device__ __constant__ float d_coefs[NU * NV] = {
     0.50f,  0.25f,  0.75f, -0.25f,
     0.30f, -0.60f,  0.10f,  0.90f,
    -0.40f,  0.20f,  0.80f, -0.70f,
     0.15f, -0.35f,  0.55f,  0.05f
};

__global__ __launch_bounds__(256)
void ftp_uuvv_wmma_kernel(const float* __restrict__ in0,
                          const float* __restrict__ in1,
                          float* __restrict__ out)
{
    // One block per (b, j); 8 waves per block, one 16-wide v-tile per wave.
    const int b    = blockIdx.x >> 2;
    const int j    = blockIdx.x & 3;
    const int wave = threadIdx.x >> 5;       // 0..7
    const int lane = threadIdx.x & 31;       // wave32 lane
    const int n    = lane & 15;              // B/D column within tile
    const int half = lane >> 4;              // 0: K=0,1   1: K=2,3
    const int v0   = wave << 4;              // v-tile base

    const float* x0 = in0 + (size_t)b * (NU * UDIM);
    const float* x1 = in1 + (size_t)b * (NU * NV * UDIM * VDIM)
                          + (size_t)j * (UDIM * VDIM);

    v8f acc = {};   // 16x16 f32 accumulator (all rows identical by construction)

    #pragma unroll
    for (int i = 0; i < NU; ++i) {
        const float c = d_coefs[i * NV + j];            // uniform -> SGPR
        const v2f cs = { c, c };

        // A source: x0[i*128 + u + 2*half + {0,1}]  (broadcast within half-wave)
        const float* aptr = x0 + i * UDIM + (half << 1);
        // B source: x1[i][u + 2*half + {0,1}][v0 + n]
        const float* bptr = x1 + (size_t)i * (NV * UDIM * VDIM)
                               + (half << 8)            // (2*half)*VDIM
                               + v0 + n;

        #pragma unroll
        for (int u = 0; u < UDIM; u += 4) {
            // A operand: K-slice of scaled x0 (b64 broadcast load, L2-resident)
            v2f a = *(const v2f*)(aptr + u);
            a = a * cs;

            // B operand: two coalesced 64B half-wave rows of in1, streamed NT
            v2f bm;
            bm.x = __builtin_nontemporal_load(bptr + u * VDIM);          // K=u+2h
            bm.y = __builtin_nontemporal_load(bptr + u * VDIM + VDIM);   // K=u+2h+1

            // D = A x B + C   (v_wmma_f32_16x16x4_f32)
            acc = __builtin_amdgcn_wmma_f32_16x16x4_f32(
                      /*neg_a=*/false, a,
                      /*neg_b=*/false, bm,
                      /*c_mod=*/(short)0, acc,
                      /*reuse_a=*/false, /*reuse_b=*/false);
        }
    }

    // Row M=0 of D lives in acc[0], lanes 0..15 (N = lane). Lanes 16..31 hold
    // the identical M=8 row -> mask them off for the store (EXEC was all-1s
    // for every WMMA above; masking only the final store is legal).
    if (half == 0) {
        out[(size_t)b * (NV * VDIM) + j * VDIM + v0 + n] = acc[0];
    }
}

extern "C" void kernel_launch(void* const* d_in, const int* in_sizes, int n_in,
                              void* d_out, int out_size, void* d_ws, size_t ws_size,
                              hipStream_t stream)
{
    const float* in0 = (const float*)d_in[0];
    const float* in1 = (const float*)d_in[1];
    float* out       = (float*)d_out;

    const int B = in_sizes[0] / (NU * UDIM);   // 512 for the reference shapes

    dim3 grid(B * NV);     // one block per (batch, output segment j)
    dim3 block(256);       // 8 wave32 waves -> 8 v-tiles of 16
    ftp_uuvv_wmma_kernel<<<grid, block, 0, stream>>>(in0, in1, out);
}